// First_n_Last_66855460929563
// MI455X (gfx1250) — compile-verified
//
#include <hip/hip_runtime.h>
#include <stdint.h>

// Problem constants (from the reference)
#define Bq      4
#define Sq      2048
#define Hq      768            // output cols (GEMM N)
#define NSPANS  24576          // S * MAX_W
#define MTOT    (Bq * NSPANS)  // 98304 GEMM rows
#define KTOT    (2 * Hq)       // 1536

// Tiling
#define BM 128
#define BN 64
#define BK 32
#define LDA 36                 // 144B row stride: 16B-aligned rows, conflict-free frag reads
#define LDB 36
#define ABUF (BM * LDA)        // floats per A buffer
#define WBUF (BN * LDB)        // floats per W buffer

typedef __attribute__((ext_vector_type(2))) float v2f;
typedef __attribute__((ext_vector_type(8))) float v8f;
typedef int v4i_vs __attribute__((__vector_size__(4 * sizeof(int)))); // matches builtin param

#define GLOBAL_AS __attribute__((address_space(1)))
#define LDS_AS    __attribute__((address_space(3)))

#if __has_builtin(__builtin_amdgcn_global_load_async_to_lds_b128)
#define HAVE_ASYNC_LDS 1
#endif

__device__ __forceinline__ void wait_asynccnt0() {
#if __has_builtin(__builtin_amdgcn_s_wait_asynccnt)
    __builtin_amdgcn_s_wait_asynccnt(0);
#else
    asm volatile("s_wait_asynccnt 0x0" ::: "memory");
#endif
}

__global__ __launch_bounds__(256)
void span_first_last_gemm(const float* __restrict__ token_reps,        // (B,S,H) f32
                          const long long* __restrict__ span_ids,      // (B,N,2) i64
                          const unsigned char* __restrict__ span_masks,// (B,N) bool
                          const float* __restrict__ Wm,                // (2H,H) f32
                          const float* __restrict__ bias,              // (H,) f32
                          float* __restrict__ out)                     // (B,N,H) f32
{
    __shared__ float A_s[2 * ABUF];        // double-buffered gathered A tiles
    __shared__ float Wt_s[2 * WBUF];       // double-buffered W tiles, transposed [n][k]
    __shared__ int   rowGather[2 * BM];    // [0:BM)=start rows, [BM:2BM)=end rows
    __shared__ unsigned char rowMask[BM];  // span validity, applied at epilogue

    const int tid  = threadIdx.x;
    const int lane = tid & 31;
    const int wave = tid >> 5;
    const int wm   = wave & 3;             // wave M position (0..3)
    const int wn   = wave >> 2;            // wave N position (0..1)
    const int g    = lane >> 4;            // half-wave group
    const int lr   = lane & 15;

    const int block_n0 = blockIdx.x * BN;  // 12 blocks over N=768
    const int block_m0 = blockIdx.y * BM;  // 768 blocks over M=98304

    // ---- Prologue: resolve gather rows; mask deferred to epilogue -----------
    // (masked span => zero A row => out = bias, so we can gather unconditionally
    //  and select at store time; this keeps the A path pure async copies)
    if (tid < BM) {
        int rg = block_m0 + tid;
        rowMask[tid] = span_masks[rg];
        long long s = span_ids[2 * (long long)rg];
        long long e = span_ids[2 * (long long)rg + 1] - 1;   // inclusive end
        int base = (rg / NSPANS) * Sq;
        rowGather[tid]      = base + (int)s;
        rowGather[BM + tid] = base + (int)e;
    }
    __syncthreads();

    const int half = tid & 1;              // which 16-col half of the A row
    const int arow = tid >> 1;             // 0..127: A-tile row this thread fills

    // A-tile loader: 32 contiguous floats per row (768 % 32 == 0, so a K-tile
    // never straddles the start/end boundary). Async global->LDS, no VGPRs.
    auto load_A = [&](int kt, int buf) {
        const int side = (kt >= Hq);
        const int ridx = rowGather[side * BM + arow];
        const int c0   = kt - side * Hq + half * 16;
        const float* gp = token_reps + (size_t)ridx * Hq + c0;
        float* lp = &A_s[buf * ABUF + arow * LDA + half * 16];
#ifdef HAVE_ASYNC_LDS
        float* gnc = const_cast<float*>(gp);
        __builtin_amdgcn_global_load_async_to_lds_b128(
            (GLOBAL_AS v4i_vs*)(gnc + 0),  (LDS_AS v4i_vs*)(lp + 0),  0, 0);
        __builtin_amdgcn_global_load_async_to_lds_b128(
            (GLOBAL_AS v4i_vs*)(gnc + 4),  (LDS_AS v4i_vs*)(lp + 4),  0, 0);
        __builtin_amdgcn_global_load_async_to_lds_b128(
            (GLOBAL_AS v4i_vs*)(gnc + 8),  (LDS_AS v4i_vs*)(lp + 8),  0, 0);
        __builtin_amdgcn_global_load_async_to_lds_b128(
            (GLOBAL_AS v4i_vs*)(gnc + 12), (LDS_AS v4i_vs*)(lp + 12), 0, 0);
#else
        const float4* gs = (const float4*)gp;
        float4* ls = (float4*)lp;                       // 16B-aligned (LDA=36)
        ls[0] = gs[0]; ls[1] = gs[1]; ls[2] = gs[2]; ls[3] = gs[3];
#endif
    };

    // W-tile: global->VGPR fetch (overlaps compute), transposed ds store later.
    auto fetch_W = [&](int kt, float4* wv) {
        #pragma unroll
        for (int i = 0; i < 2; ++i) {
            int l  = tid + i * 256;                     // 512 float4s total
            int kr = l >> 4;                            // K row in tile
            int nc = (l & 15) << 2;                     // N col group
            wv[i] = *(const float4*)(Wm + (size_t)(kt + kr) * Hq + block_n0 + nc);
        }
    };
    auto store_W = [&](int buf, const float4* wv) {
        float* base = &Wt_s[buf * WBUF];
        #pragma unroll
        for (int i = 0; i < 2; ++i) {
            int l  = tid + i * 256;
            int kr = l >> 4;
            int nc = (l & 15) << 2;
            base[(nc + 0) * LDB + kr] = wv[i].x;
            base[(nc + 1) * LDB + kr] = wv[i].y;
            base[(nc + 2) * LDB + kr] = wv[i].z;
            base[(nc + 3) * LDB + kr] = wv[i].w;
        }
    };

    v8f acc[2][2] = {};
    float4 wv[2];

    // ---- Prime buffer 0 ----
    load_A(0, 0);
    fetch_W(0, wv);
    store_W(0, wv);
    wait_asynccnt0();
    __syncthreads();

    constexpr int NT = KTOT / BK;          // 48 K-tiles
    for (int t = 0; t < NT; ++t) {
        const int cur = t & 1;
        const int nxt = cur ^ 1;
        const bool has_next = (t + 1) < NT;

        // Kick off next tile's loads: async A writes straight to LDS buf[nxt],
        // W rides in VGPRs — both overlap the WMMA stream below.
        if (has_next) {
            load_A((t + 1) * BK, nxt);
            fetch_W((t + 1) * BK, wv);
        }

        // ---- Compute on buf[cur]: 8 k-steps x 4 WMMAs -----------------------
        // fp32 WMMA layout: lane L supplies A[M=L%16, K=2*(L/16)+j] and
        // B[K=2*(L/16)+j, N=L%16], j=0,1 -> one b64 LDS read per fragment.
        const float* Ab  = &A_s[cur * ABUF];
        const float* Bb  = &Wt_s[cur * WBUF];
        const float* Ab0 = Ab + (wm * 32 +  0 + lr) * LDA;
        const float* Ab1 = Ab + (wm * 32 + 16 + lr) * LDA;
        const float* Bb0 = Bb + (wn * 32 +  0 + lr) * LDB;
        const float* Bb1 = Bb + (wn * 32 + 16 + lr) * LDB;
        #pragma unroll
        for (int ks = 0; ks < BK; ks += 4) {
            const int ko = ks + 2 * g;
            v2f a0 = *(const v2f*)(Ab0 + ko);
            v2f a1 = *(const v2f*)(Ab1 + ko);
            v2f b0 = *(const v2f*)(Bb0 + ko);
            v2f b1 = *(const v2f*)(Bb1 + ko);
            acc[0][0] = __builtin_amdgcn_wmma_f32_16x16x4_f32(
                false, a0, false, b0, (short)0, acc[0][0], false, false);
            acc[0][1] = __builtin_amdgcn_wmma_f32_16x16x4_f32(
                false, a0, false, b1, (short)0, acc[0][1], false, false);
            acc[1][0] = __builtin_amdgcn_wmma_f32_16x16x4_f32(
                false, a1, false, b0, (short)0, acc[1][0], false, false);
            acc[1][1] = __builtin_amdgcn_wmma_f32_16x16x4_f32(
                false, a1, false, b1, (short)0, acc[1][1], false, false);
        }

        // Commit next W tile to LDS, then close the pipeline stage:
        // __syncthreads covers DScnt; asynccnt needs an explicit wait.
        if (has_next) store_W(nxt, wv);
        wait_asynccnt0();
        __syncthreads();
    }

    // ---- Epilogue: mask + bias, store (C/D layout: VGPR v -> M = v + 8*g) ---
    #pragma unroll
    for (int tn = 0; tn < 2; ++tn) {
        const int col = block_n0 + wn * 32 + tn * 16 + lr;
        const float bv = bias[col];
        #pragma unroll
        for (int tm = 0; tm < 2; ++tm) {
            const int lrow = wm * 32 + tm * 16 + 8 * g;
            #pragma unroll
            for (int v = 0; v < 8; ++v) {
                const int r = lrow + v;
                const float val = rowMask[r] ? (acc[tm][tn][v] + bv) : bv;
                out[(size_t)(block_m0 + r) * Hq + col] = val;
            }
        }
    }
}

extern "C" void kernel_launch(void* const* d_in, const int* in_sizes, int n_in,
                              void* d_out, int out_size, void* d_ws, size_t ws_size,
                              hipStream_t stream) {
    const float*         token_reps = (const float*)d_in[0];
    const long long*     span_ids   = (const long long*)d_in[1];     // int64 in reference
    const unsigned char* span_masks = (const unsigned char*)d_in[2]; // bool
    const float*         Wm         = (const float*)d_in[3];
    const float*         bias       = (const float*)d_in[4];
    float*               out        = (float*)d_out;

    dim3 grid(Hq / BN, MTOT / BM);   // (12, 768)
    span_first_last_gemm<<<grid, 256, 0, stream>>>(
        token_reps, span_ids, span_masks, Wm, bias, out);
}